// OrthogonalLayer_46480136077864
// MI455X (gfx1250) — compile-verified
//
#include <hip/hip_runtime.h>

typedef __attribute__((ext_vector_type(2))) float v2f;
typedef __attribute__((ext_vector_type(8))) float v8f;

#define NROW 64
#define NF   256
#define LDA  258   // pad 64x256 rows by 2 floats: rows land on different banks, keeps 8B alignment
#define LDG  66    // pad 64x64

__global__ __launch_bounds__(256)
void cholqr_gs_kernel(const float* __restrict__ feats, float* __restrict__ out) {
    __shared__ float sA[NROW][LDA];   // input rows (row-major)
    __shared__ float sG[NROW][LDG];   // Gram matrix -> Cholesky factor L (lower)
    __shared__ float sW[NROW][LDG];   // W = L^{-1} (lower; upper zeroed)

    const int tid  = threadIdx.x;
    const int lane = tid & 31;
    const int wave = tid >> 5;
    const int lm   = lane & 15;          // M (A-frag) or N (B-frag / C-frag col)
    const int lk   = (lane >> 4) << 1;   // K base: 0 or 2
    const int mh   = (lane >> 4) << 3;   // C/D row base: 0 or 8
    const int b    = blockIdx.x;

    // ---------------- Phase 1: load feats[b] (64x256 f32) into LDS ----------------
    {
        const float4* src = (const float4*)(feats + (size_t)b * (NROW * NF));
        #pragma unroll
        for (int i = 0; i < 16; ++i) {
            int idx = tid + i * 256;          // 4096 float4 per batch
            float4 x = src[idx];
            int r = idx >> 6;                 // 64 float4 per row
            int c = (idx & 63) << 2;
            sA[r][c + 0] = x.x; sA[r][c + 1] = x.y;
            sA[r][c + 2] = x.z; sA[r][c + 3] = x.w;
        }
    }
    // zero W while loads are in flight (upper triangle must be 0 for phase 4)
    for (int i = tid; i < NROW * LDG; i += 256) ((float*)sW)[i] = 0.f;
    __syncthreads();

    // ---------------- Phase 2: G = A * A^T (lower-triangle tiles) via WMMA ----------------
    {
        const int ti_tab[10] = {0,1,1,2,2,2,3,3,3,3};
        const int tj_tab[10] = {0,0,1,0,1,2,0,1,2,3};
        for (int t = wave; t < 10; t += 8) {
            const int ti = ti_tab[t], tj = tj_tab[t];
            v8f acc = {0.f,0.f,0.f,0.f,0.f,0.f,0.f,0.f};
            for (int k0 = 0; k0 < NF; k0 += 4) {
                v2f a, bb;
                a.x  = sA[ti*16 + lm][k0 + lk];
                a.y  = sA[ti*16 + lm][k0 + lk + 1];
                bb.x = sA[tj*16 + lm][k0 + lk];      // B[k][n] = A[tj*16+n][k]
                bb.y = sA[tj*16 + lm][k0 + lk + 1];
                acc = __builtin_amdgcn_wmma_f32_16x16x4_f32(
                        false, a, false, bb, (short)0, acc, false, false);
            }
            #pragma unroll
            for (int v = 0; v < 8; ++v)
                sG[ti*16 + mh + v][tj*16 + lm] = acc[v];
        }
    }

    // per-thread fixed ownership of the 2080-entry 64x64 lower triangle (incl. diagonal)
    int pr[9], pc[9];
    #pragma unroll
    for (int j = 0; j < 9; ++j) {
        int p = tid + j * 256;
        if (p < 2080) {
            int r = (int)((sqrtf(8.f * (float)p + 1.f) - 1.f) * 0.5f);
            while ((r + 1) * (r + 2) / 2 <= p) ++r;
            while (r * (r + 1) / 2 > p) --r;
            pr[j] = r; pc[j] = p - r * (r + 1) / 2;
        } else {
            pr[j] = 0; pc[j] = 0;   // (0,0): condition c>k never fires -> inert
        }
    }
    __syncthreads();

    // ---------------- Phase 3a: right-looking Cholesky of G (in place, lower) ----------------
    for (int k = 0; k < NROW; ++k) {
        if (tid == 0) {
            float g = sG[k][k];
            float d = sqrtf(fmaxf(g, 0.f));
            sG[k][k] = fmaxf(d, 1e-8f);       // mirrors max(norm, EPS)
        }
        __syncthreads();
        float invd = 1.f / sG[k][k];
        int r = k + 1 + tid;
        if (r < NROW) sG[r][k] *= invd;
        __syncthreads();
        #pragma unroll
        for (int j = 0; j < 9; ++j) {
            int rr = pr[j], cc = pc[j];
            if (cc > k) sG[rr][cc] -= sG[rr][k] * sG[cc][k];
        }
        __syncthreads();
    }

    // ---------------- Phase 3b: W = L^{-1}, thread-per-column forward substitution ----------------
    if (tid < NROW) {
        const int c = tid;
        sW[c][c] = 1.f / sG[c][c];
        for (int r = c + 1; r < NROW; ++r) {
            float s = 0.f;
            for (int j = c; j < r; ++j) s += sG[r][j] * sW[j][c];
            sW[r][c] = -s / sG[r][r];
        }
    }
    __syncthreads();

    // ---------------- Phase 4: Q = W * A via WMMA; stream tiles to global ----------------
    for (int t = wave; t < 64; t += 8) {
        const int rp = t >> 4;                 // row panel 0..3
        const int ct = (t & 15) << 4;          // column base 0..240
        const int kend = (rp + 1) << 4;        // W is lower-triangular: truncate K
        v8f acc = {0.f,0.f,0.f,0.f,0.f,0.f,0.f,0.f};
        for (int k0 = 0; k0 < kend; k0 += 4) {
            v2f a, bb;
            a.x  = sW[rp*16 + lm][k0 + lk];
            a.y  = sW[rp*16 + lm][k0 + lk + 1];
            bb.x = sA[k0 + lk][ct + lm];       // B[k][n] = A[k][ct+n]
            bb.y = sA[k0 + lk + 1][ct + lm];
            acc = __builtin_amdgcn_wmma_f32_16x16x4_f32(
                    false, a, false, bb, (short)0, acc, false, false);
        }
        float* op = out + (size_t)b * (NROW * NF) + (size_t)(rp * 16) * NF + ct;
        #pragma unroll
        for (int v = 0; v < 8; ++v)
            op[(mh + v) * NF + lm] = acc[v];
    }
}

extern "C" void kernel_launch(void* const* d_in, const int* in_sizes, int n_in,
                              void* d_out, int out_size, void* d_ws, size_t ws_size,
                              hipStream_t stream) {
    const float* feats = (const float*)d_in[0];
    float* out = (float*)d_out;
    const int B = in_sizes[0] / (NROW * NF);   // 4096
    hipLaunchKernelGGL(cholqr_gs_kernel, dim3(B), dim3(256), 0, stream, feats, out);
}